// MOGConv_14963666059791
// MI455X (gfx1250) — compile-verified
//
#include <hip/hip_runtime.h>
#include <math.h>

// ---------------------------------------------------------------------------
// MOGConv forward for MI455X (gfx1250), fp32 throughout.
// All GEMMs run on the fp32 matrix pipe via V_WMMA_F32_16X16X4_F32 with a
// depth-1 software pipeline (next K-step loads overlap current WMMAs).
// Feature maps stored channel-major (C, N); edge features stored as the
// gathered (E, C) buffer reinterpreted as a (C, E) matrix (ld = E), which
// reproduces the reference's raw .view exactly.
// ---------------------------------------------------------------------------

typedef float v2f __attribute__((ext_vector_type(2)));
typedef float v8f __attribute__((ext_vector_type(8)));

#define NPTS   8192LL       // bs * n
#define HID    128LL
#define KNN    20LL
#define NEDGE  (NPTS * KNN) // 163840
#define CDIM_  128LL
#define HDR    24576LL      // bs*n*3 floats of pc copied to d_out first
#define NT     4            // 16-col tiles per wave (64 output columns)

// ---------------- small helpers ----------------

__device__ __forceinline__ unsigned enc_ord(float f) {
  unsigned u = __float_as_uint(f);
  return (u & 0x80000000u) ? ~u : (u | 0x80000000u);
}
__device__ __forceinline__ float dec_ord(unsigned u) {
  return __uint_as_float((u & 0x80000000u) ? (u ^ 0x80000000u) : ~u);
}
#define ENC_NEGINF 0x007FFFFFu   // enc_ord(-inf)

// pc (N,3) row-major -> P3 (3,N)
__global__ void k_transpose3(const float* __restrict__ pc, float* __restrict__ p3) {
  long long tid = (long long)blockIdx.x * blockDim.x + threadIdx.x;
  if (tid >= 3 * NPTS) return;
  long long c = tid / NPTS, n = tid % NPTS;
  p3[c * NPTS + n] = pc[n * 3 + c];
}

// (C,N) -> (N,C) tiled transpose (for the coalesced gather source)
__global__ __launch_bounds__(256)
void k_transpose_cn(const float* __restrict__ X, float* __restrict__ Xt, int C) {
  __shared__ float tile[32][33];
  const int n0 = blockIdx.x * 32, c0 = blockIdx.y * 32;
  for (int r = threadIdx.y; r < 32; r += 8)
    tile[r][threadIdx.x] = X[(long long)(c0 + r) * NPTS + n0 + threadIdx.x];
  __syncthreads();
  for (int r = threadIdx.y; r < 32; r += 8)
    Xt[(long long)(n0 + r) * C + c0 + threadIdx.x] = tile[threadIdx.x][r];
}

// sq[j] = sum_c X[c*N+j]^2 , X is (C,N)
__global__ void k_sqnorm(const float* __restrict__ X, int C, float* __restrict__ sq) {
  long long j = (long long)blockIdx.x * blockDim.x + threadIdx.x;
  if (j >= NPTS) return;
  float s = 0.f;
  for (int c = 0; c < C; ++c) { float v = X[(long long)c * NPTS + j]; s = fmaf(v, v, s); }
  sq[j] = s;
}

// KNN within same tag, excluding self; one block (128 thr) per query point.
// Emits indices sorted by ascending distance (ties -> lower index), matching
// jax.lax.top_k(-d, k).
__global__ __launch_bounds__(128)
void k_knn(const float* __restrict__ X, int C, const float* __restrict__ sq,
           const int* __restrict__ tag, int* __restrict__ knn) {
  const int i = blockIdx.x;
  const int t = threadIdx.x;
  __shared__ float qf[128];
  __shared__ float lv[128 * 20];
  __shared__ int   li[128 * 20];
  __shared__ float rv[128];
  __shared__ int   ri[128];
  __shared__ int   rt[128];
  if (t < C) qf[t] = X[(long long)t * NPTS + i];
  __syncthreads();
  const float qsq = sq[i];
  const int myTag = tag[i];
  float* Lv = lv + t * 20;
  int*   Li = li + t * 20;
  int cnt = 0;
  for (long long j = t; j < NPTS; j += 128) {
    if (j == i || tag[j] != myTag) continue;
    float dot = 0.f;
    for (int c = 0; c < C; ++c) dot = fmaf(qf[c], X[(long long)c * NPTS + j], dot);
    float d = qsq + sq[j] - 2.0f * dot;
    int ji = (int)j;
    if (cnt < 20) {
      int p = cnt++;
      while (p > 0 && (Lv[p - 1] > d || (Lv[p - 1] == d && Li[p - 1] > ji))) {
        Lv[p] = Lv[p - 1]; Li[p] = Li[p - 1]; --p;
      }
      Lv[p] = d; Li[p] = ji;
    } else if (d < Lv[19] || (d == Lv[19] && ji < Li[19])) {
      int p = 19;
      while (p > 0 && (Lv[p - 1] > d || (Lv[p - 1] == d && Li[p - 1] > ji))) {
        Lv[p] = Lv[p - 1]; Li[p] = Li[p - 1]; --p;
      }
      Lv[p] = d; Li[p] = ji;
    }
  }
  __syncthreads();
  // merge 128 sorted lists -> global top-20
  int p = 0;
  for (int r = 0; r < 20; ++r) {
    float v = (p < cnt) ? Lv[p] : INFINITY;
    int  id = (p < cnt) ? Li[p] : 0x7fffffff;
    rv[t] = v; ri[t] = id; rt[t] = t;
    __syncthreads();
    for (int s = 64; s > 0; s >>= 1) {
      if (t < s) {
        float v2 = rv[t + s]; int i2 = ri[t + s];
        if (v2 < rv[t] || (v2 == rv[t] && i2 < ri[t])) {
          rv[t] = v2; ri[t] = i2; rt[t] = rt[t + s];
        }
      }
      __syncthreads();
    }
    if (t == 0) knn[(long long)i * 20 + r] = (ri[0] == 0x7fffffff) ? 0 : ri[0];
    int wt = rt[0];
    __syncthreads();
    if (t == wt) ++p;
    __syncthreads();
  }
}

// G[e*C + c] = Xt[idx[e]*C + c]  (Xt is (N,C) row-major; both sides coalesced)
__global__ void k_gather(const float* __restrict__ Xt, int C,
                         const int* __restrict__ idx, float* __restrict__ G) {
  long long tid = (long long)blockIdx.x * blockDim.x + threadIdx.x;
  long long total = NEDGE * (long long)C;
  if (tid >= total) return;
  long long e = tid / C;
  int c = (int)(tid % C);
  G[tid] = Xt[(long long)idx[e] * C + c];
}

// ---------------- WMMA GEMM ----------------

__device__ __forceinline__ v8f wmma4(v2f av, v2f bv, v8f acc) {
  return __builtin_amdgcn_wmma_f32_16x16x4_f32(false, av, false, bv,
                                               (short)0, acc, false, false);
}

// One contiguous B segment, depth-1 software pipeline: the loads for step s+1
// are issued before the WMMAs of step s, so memory latency overlaps matrix ops.
__device__ __forceinline__ void gemm_seg(const float*& a, const float* b,
                                         long long Ncols, int steps, v8f* acc) {
  if (steps <= 0) return;
  float a0 = a[0], a1 = a[1];
  float b0[NT], b1[NT];
#pragma unroll
  for (int t = 0; t < NT; ++t) { b0[t] = b[t * 16]; b1[t] = b[t * 16 + Ncols]; }
  for (int s = 0; s < steps - 1; ++s) {
    a += 4; b += 4 * Ncols;
    float a0n = a[0], a1n = a[1];               // prefetch next K-step
    float b0n[NT], b1n[NT];
#pragma unroll
    for (int t = 0; t < NT; ++t) { b0n[t] = b[t * 16]; b1n[t] = b[t * 16 + Ncols]; }
    v2f av; av[0] = a0; av[1] = a1;
#pragma unroll
    for (int t = 0; t < NT; ++t) {
      v2f bv; bv[0] = b0[t]; bv[1] = b1[t];
      acc[t] = wmma4(av, bv, acc[t]);
    }
    a0 = a0n; a1 = a1n;
#pragma unroll
    for (int t = 0; t < NT; ++t) { b0[t] = b0n[t]; b1[t] = b1n[t]; }
  }
  a += 4;
  v2f av; av[0] = a0; av[1] = a1;
#pragma unroll
  for (int t = 0; t < NT; ++t) {
    v2f bv; bv[0] = b0[t]; bv[1] = b1[t];
    acc[t] = wmma4(av, bv, acc[t]);
  }
}

// Y(M x Ncols) = W(M x Kd) @ B(Kd x Ncols); B is two stacked row segments:
// rows [0,C1) from B1, rows [C1,Kd) from B2, each row-major with ld = Ncols.
// One wave per 16x64 output tile (4 WMMAs per K-step, A fragment reused 4x).
__global__ __launch_bounds__(256)
void k_gemm_wmma(const float* __restrict__ Wm, int M, int Kd,
                 const float* __restrict__ B1, int C1,
                 const float* __restrict__ B2,
                 float* __restrict__ Y, long long Ncols) {
  const int lane = threadIdx.x & 31;
  const int wave = threadIdx.x >> 5;
  long long tile = (long long)blockIdx.x * (blockDim.x >> 5) + wave;
  const long long ntn = Ncols >> 6;                 // 64-column supertiles
  const long long nTiles = (long long)(M >> 4) * ntn;
  if (tile >= nTiles) return;
  const int m0 = (int)(tile / ntn) << 4;
  const long long n0 = (tile % ntn) << 6;
  const int lo = lane & 15, hi = lane >> 4;

  v8f zero = {0.f, 0.f, 0.f, 0.f, 0.f, 0.f, 0.f, 0.f};
  v8f acc[NT];
#pragma unroll
  for (int t = 0; t < NT; ++t) acc[t] = zero;

  const int KA = (C1 < Kd) ? C1 : Kd;
  const int steps1 = KA >> 2;                       // full steps from B1
  const int steps2 = (Kd >> 2) - steps1;            // full steps from B2

  const float* a = Wm + (long long)(m0 + lo) * Kd + 2 * hi;
  gemm_seg(a, B1 + (long long)(2 * hi) * Ncols + n0 + lo, Ncols, steps1, acc);
  gemm_seg(a, B2 + (long long)(2 * hi) * Ncols + n0 + lo, Ncols, steps2, acc);

  // cold masked tail (only Kd % 4 != 0, i.e. the Kd=3 first conv; C1 == Kd there)
  int k = (steps1 + steps2) << 2;
  if (k < Kd) {
    float a0 = 0.f, a1 = 0.f;
    float b0[NT] = {0.f, 0.f, 0.f, 0.f}, b1[NT] = {0.f, 0.f, 0.f, 0.f};
    const int c0 = k + 2 * hi, c1 = c0 + 1;
    if (c0 < Kd) {
      a0 = Wm[(long long)(m0 + lo) * Kd + c0];
      const float* r0 = ((c0 < C1) ? B1 + (long long)c0 * Ncols
                                   : B2 + (long long)(c0 - C1) * Ncols) + n0 + lo;
#pragma unroll
      for (int t = 0; t < NT; ++t) b0[t] = r0[t * 16];
    }
    if (c1 < Kd) {
      a1 = Wm[(long long)(m0 + lo) * Kd + c1];
      const float* r1 = ((c1 < C1) ? B1 + (long long)c1 * Ncols
                                   : B2 + (long long)(c1 - C1) * Ncols) + n0 + lo;
#pragma unroll
      for (int t = 0; t < NT; ++t) b1[t] = r1[t * 16];
    }
#pragma unroll
    for (int t = 0; t < NT; ++t) {
      v2f av; av[0] = a0; av[1] = a1;
      v2f bv; bv[0] = b0[t]; bv[1] = b1[t];
      acc[t] = wmma4(av, bv, acc[t]);
    }
  }

#pragma unroll
  for (int t = 0; t < NT; ++t)
#pragma unroll
    for (int r = 0; r < 8; ++r)
      Y[(long long)(m0 + r + 8 * hi) * Ncols + n0 + t * 16 + lo] = acc[t][r];
}

// per-row mean / rstd over L columns (biased variance + eps, torch InstanceNorm)
__global__ __launch_bounds__(256)
void k_rowstat(const float* __restrict__ Y, long long L,
               float* __restrict__ mean, float* __restrict__ rstd) {
  const int r = blockIdx.x;
  const float* row = Y + (long long)r * L;
  float s = 0.f, s2 = 0.f;
  for (long long i = threadIdx.x; i < L; i += blockDim.x) {
    float v = row[i]; s += v; s2 = fmaf(v, v, s2);
  }
  __shared__ float sh[256], sh2[256];
  sh[threadIdx.x] = s; sh2[threadIdx.x] = s2; __syncthreads();
  for (int st = 128; st > 0; st >>= 1) {
    if (threadIdx.x < st) { sh[threadIdx.x] += sh[threadIdx.x + st]; sh2[threadIdx.x] += sh2[threadIdx.x + st]; }
    __syncthreads();
  }
  if (threadIdx.x == 0) {
    float m = sh[0] / (float)L;
    float v = sh2[0] / (float)L - m * m;
    if (v < 0.f) v = 0.f;
    mean[r] = m; rstd[r] = rsqrtf(v + 1e-5f);
  }
}

// in-place y = lrelu((y - mean[c]) * rstd[c]), y is (M, L)
__global__ void k_normact(float* __restrict__ Y, int M, long long L,
                          const float* __restrict__ mean, const float* __restrict__ rstd) {
  long long tid = (long long)blockIdx.x * blockDim.x + threadIdx.x;
  if (tid >= (long long)M * L) return;
  int c = (int)(tid / L);
  float v = (Y[tid] - mean[c]) * rstd[c];
  Y[tid] = (v >= 0.f) ? v : 0.2f * v;
}

__global__ void k_fill_u32(unsigned* __restrict__ p, long long n, unsigned v) {
  long long tid = (long long)blockIdx.x * blockDim.x + threadIdx.x;
  if (tid < n) p[tid] = v;
}

// ordered-uint atomic segment max: agg[h*N + idx[e]] max= Y[h*E + e]
__global__ void k_segmax(const float* __restrict__ Y, const int* __restrict__ idx,
                         unsigned* __restrict__ agg) {
  long long tid = (long long)blockIdx.x * blockDim.x + threadIdx.x;
  if (tid >= HID * NEDGE) return;
  long long h = tid / NEDGE, e = tid % NEDGE;
  atomicMax(&agg[h * NPTS + idx[e]], enc_ord(Y[tid]));
}

// decode, -inf -> 0, add residual (prev may be null), write (H,N) feature map
__global__ void k_segfix(const unsigned* __restrict__ agg, const float* __restrict__ prev,
                         float* __restrict__ Xout) {
  long long tid = (long long)blockIdx.x * blockDim.x + threadIdx.x;
  if (tid >= HID * NPTS) return;
  float f = dec_ord(agg[tid]);
  if (isinf(f) && f < 0.f) f = 0.f;
  Xout[tid] = f + (prev ? prev[tid] : 0.f);
}

// per-row max over N then broadcast back into the row (g = broadcast(max(g,1)))
__global__ __launch_bounds__(256)
void k_rowmax_bcast(float* __restrict__ G, long long L) {
  const int r = blockIdx.x;
  float* row = G + (long long)r * L;
  float m = -INFINITY;
  for (long long i = threadIdx.x; i < L; i += blockDim.x) m = fmaxf(m, row[i]);
  __shared__ float sh[256];
  sh[threadIdx.x] = m; __syncthreads();
  for (int st = 128; st > 0; st >>= 1) {
    if (threadIdx.x < st) sh[threadIdx.x] = fmaxf(sh[threadIdx.x], sh[threadIdx.x + st]);
    __syncthreads();
  }
  float gm = sh[0];
  for (long long i = threadIdx.x; i < L; i += blockDim.x) row[i] = gm;
}

// out[HDR + n*CDIM + cd] = T[cd*N + n] + br[cd]
__global__ void k_final(const float* __restrict__ T, const float* __restrict__ br,
                        float* __restrict__ out) {
  long long tid = (long long)blockIdx.x * blockDim.x + threadIdx.x;
  if (tid >= NPTS * CDIM_) return;
  long long n = tid / CDIM_, cd = tid % CDIM_;
  out[HDR + tid] = T[cd * NPTS + n] + br[cd];
}

__global__ void k_copy(const float* __restrict__ src, float* __restrict__ dst, long long n) {
  long long tid = (long long)blockIdx.x * blockDim.x + threadIdx.x;
  if (tid < n) dst[tid] = src[tid];
}

// ---------------------------------------------------------------------------

static inline int gblk(long long total, int thr) { return (int)((total + thr - 1) / thr); }

static void run_gemm(const float* Wm, int M, int Kd, const float* B1, int C1,
                     const float* B2, float* Y, long long Ncols, hipStream_t s) {
  long long tiles = (long long)(M >> 4) * (Ncols >> 6);   // 16x64 tiles per wave
  k_gemm_wmma<<<gblk(tiles, 8), 256, 0, s>>>(Wm, M, Kd, B1, C1, B2 ? B2 : B1, Y, Ncols);
}

static void run_normblock(float* Y, int M, long long L, float* mean, float* rstd, hipStream_t s) {
  k_rowstat<<<M, 256, 0, s>>>(Y, L, mean, rstd);
  k_normact<<<gblk((long long)M * L, 256), 256, 0, s>>>(Y, M, L, mean, rstd);
}

extern "C" void kernel_launch(void* const* d_in, const int* in_sizes, int n_in,
                              void* d_out, int out_size, void* d_ws, size_t ws_size,
                              hipStream_t stream) {
  const float* pc  = (const float*)d_in[0];
  const int*   tag = (const int*)d_in[1];
  const float* Wab[8];
  for (int i = 0; i < 8; ++i) Wab[i] = (const float*)d_in[2 + i];
  const float* W9  = (const float*)d_in[10];
  const float* W10 = (const float*)d_in[11];
  const float* W11 = (const float*)d_in[12];
  const float* W12 = (const float*)d_in[13];
  const float* W13 = (const float*)d_in[14];
  const float* Wr  = (const float*)d_in[15];
  const float* br  = (const float*)d_in[16];
  float* out = (float*)d_out;
  char* ws = (char*)d_ws;

  // ---- workspace layout (bytes) ----
  size_t oP3   = 0;
  size_t oSQ   = oP3  + (size_t)(3 * NPTS) * 4;
  size_t oIDX  = oSQ  + (size_t)NPTS * 4;
  size_t oMEAN = oIDX + (size_t)NEDGE * 4;
  size_t oRSTD = oMEAN + 512 * 4;
  size_t oXT   = oRSTD + 512 * 4;                         // (N,C) transposed gather source
  size_t oXF   = oXT + (size_t)(HID * NPTS) * 4;          // x1..x4 contiguous (4H x N)
  size_t oBIGA = oXF + (size_t)(4 * HID * NPTS) * 4;
  size_t oBIGB = oBIGA + (size_t)(NEDGE * HID) * 4;
  // total = oBIGB + NEDGE*HID*4  (~181 MB)

  float* P3   = (float*)(ws + oP3);
  float* SQ   = (float*)(ws + oSQ);
  int*   IDX  = (int*)(ws + oIDX);
  float* MEAN = (float*)(ws + oMEAN);
  float* RSTD = (float*)(ws + oRSTD);
  float* XT   = (float*)(ws + oXT);
  float* XF   = (float*)(ws + oXF);    // x_i = XF + i*H*N, (H,N) each
  float* BIGA = (float*)(ws + oBIGA);
  float* BIGB = (float*)(ws + oBIGB);

  // pc (N,3) -> P3 (3,N)
  k_transpose3<<<gblk(3 * NPTS, 256), 256, 0, stream>>>(pc, P3);

  // ---- 4 edge blocks ----
  for (int blk = 0; blk < 4; ++blk) {
    const float* X = (blk == 0) ? P3 : (XF + (size_t)(blk - 1) * HID * NPTS);
    const int C = (blk == 0) ? 3 : (int)HID;
    float* Xout = XF + (size_t)blk * HID * NPTS;
    const float* prev = (blk == 0) ? nullptr : X;

    k_sqnorm<<<gblk(NPTS, 256), 256, 0, stream>>>(X, C, SQ);
    k_knn<<<(int)NPTS, 128, 0, stream>>>(X, C, SQ, tag, IDX);

    // gather source in (N,C) layout: pc is already (N,3); else transpose
    const float* XtSrc = pc;
    if (blk != 0) {
      dim3 tg((unsigned)(NPTS / 32), (unsigned)(HID / 32));
      k_transpose_cn<<<tg, dim3(32, 8), 0, stream>>>(X, XT, (int)HID);
      XtSrc = XT;
    }
    k_gather<<<gblk(NEDGE * C, 256), 256, 0, stream>>>(XtSrc, C, IDX, BIGA);

    // conv1: (H x C) @ (C x E) -> BIGB
    run_gemm(Wab[2 * blk + 0], (int)HID, C, BIGA, C, nullptr, BIGB, NEDGE, stream);
    run_normblock(BIGB, (int)HID, NEDGE, MEAN, RSTD, stream);
    // conv2: (H x H) @ (H x E) -> BIGA (overwrites gather buf)
    run_gemm(Wab[2 * blk + 1], (int)HID, (int)HID, BIGB, (int)HID, nullptr, BIGA, NEDGE, stream);
    run_normblock(BIGA, (int)HID, NEDGE, MEAN, RSTD, stream);

    // segment max (into BIGB reused as ordered-uint agg), fixup + residual
    unsigned* AGG = (unsigned*)BIGB;
    k_fill_u32<<<gblk(HID * NPTS, 256), 256, 0, stream>>>(AGG, HID * NPTS, ENC_NEGINF);
    k_segmax<<<gblk(HID * NEDGE, 256), 256, 0, stream>>>(BIGA, IDX, AGG);
    k_segfix<<<gblk(HID * NPTS, 256), 256, 0, stream>>>(AGG, prev, Xout);
  }

  // ---- encoder / decoder (all feature maps are (C,N) row-major) ----
  float* g  = BIGA;                          // 2H x N
  float* y4 = BIGA + 2 * HID * NPTS;         // 2H x N
  float* y3 = BIGA + 4 * HID * NPTS;         // 2H x N
  float* y2 = BIGA + 6 * HID * NPTS;         //  H x N
  float* y1 = BIGA + 7 * HID * NPTS;         //  H x N
  float* T  = BIGA + 8 * HID * NPTS;         // CDIM x N
  float* x1 = XF;
  float* x2 = XF + 1 * HID * NPTS;
  float* x3 = XF + 2 * HID * NPTS;
  float* x4 = XF + 3 * HID * NPTS;

  // g = IN/lrelu(W9 @ [x1;x2;x3;x4]) then broadcast row-max
  run_gemm(W9, 256, 512, XF, 512, nullptr, g, NPTS, stream);
  run_normblock(g, 256, NPTS, MEAN, RSTD, stream);
  k_rowmax_bcast<<<256, 256, 0, stream>>>(g, NPTS);

  run_gemm(W10, 256, 384, g,  256, x4, y4, NPTS, stream);
  run_normblock(y4, 256, NPTS, MEAN, RSTD, stream);
  run_gemm(W11, 256, 384, y4, 256, x3, y3, NPTS, stream);
  run_normblock(y3, 256, NPTS, MEAN, RSTD, stream);
  run_gemm(W12, 128, 384, y3, 256, x2, y2, NPTS, stream);
  run_normblock(y2, 128, NPTS, MEAN, RSTD, stream);
  run_gemm(W13, 128, 256, y2, 128, x1, y1, NPTS, stream);
  run_normblock(y1, 128, NPTS, MEAN, RSTD, stream);

  // code = (Wr @ y1 + br).T ; outputs = [pc | code]
  run_gemm(Wr, (int)CDIM_, (int)HID, y1, (int)HID, nullptr, T, NPTS, stream);
  k_copy<<<gblk(HDR, 256), 256, 0, stream>>>(pc, out, HDR);
  k_final<<<gblk(NPTS * CDIM_, 256), 256, 0, stream>>>(T, br, out);
}